// GAE_52767968199016
// MI455X (gfx1250) — compile-verified
//
#include <hip/hip_runtime.h>

#define GAMMA  0.99f
#define LAMBDA 0.97f
#define GL     (GAMMA * LAMBDA)

static constexpr int T      = 1024;
static constexpr int BATCH  = 16384;
static constexpr int CHUNK  = 64;            // timesteps handled per wave in pass 1
static constexpr int NCHUNK = T / CHUNK;     // 16
static constexpr int COLT   = BATCH / 16;    // 1024 column tiles of width 16

typedef float v2f __attribute__((ext_vector_type(2)));
typedef float v8f __attribute__((ext_vector_type(8)));

// ---------------------------------------------------------------------------
// Compile-time table of (gamma*lambda)^n, n = 0..64. Replaces all runtime
// exp2f/log2f transcendental chains with constant-memory lookups.
// ---------------------------------------------------------------------------
struct GlTab { float v[CHUNK + 1]; };

static constexpr GlTab make_tab() {
    GlTab t{};
    double p = 1.0;
    for (int i = 0; i <= CHUNK; ++i) {
        t.v[i] = (float)p;
        p *= (double)GL;
    }
    return t;
}

__constant__ GlTab GLT = make_tab();

static constexpr float GL64 = []() constexpr {
    double p = 1.0;
    for (int i = 0; i < CHUNK; ++i) p *= (double)GL;
    return (float)p;
}();

// ---------------------------------------------------------------------------
// Pass 1: one wave per (chunk, 16-column tile). Computes the chunk-local GAE
// scan exactly (64 timesteps) using 16x16 triangular matmuls on the WMMA unit:
//   local[m][n] = sum_{k>=m} gl^{k-m} * delta[k][n]   (per 16-row sub-tile)
// with sub-tile carries propagated in-register via __shfl. Writes chunk-local
// adv to out[] and the chunk head adv (t = chunk*64) to heads[].
// ---------------------------------------------------------------------------
__global__ __launch_bounds__(256) void gae_pass1(const float* __restrict__ value,
                                                 const float* __restrict__ reward,
                                                 float* __restrict__ out,
                                                 float* __restrict__ heads)
{
    const int lane    = threadIdx.x & 31;
    const int waveId  = (int)((blockIdx.x * blockDim.x + threadIdx.x) >> 5);
    const int colTile = waveId & (COLT - 1);
    const int chunk   = waveId >> 10;          // COLT == 1024
    const int n       = lane & 15;             // column within tile / A-row m
    const int hi      = lane >> 4;             // half-wave select
    const int b0      = colTile * 16;

    // A-matrix = U (16x16 upper triangular, U[m][k] = gl^(k-m)), K sliced by 4.
    // 32-bit A 16x4 layout: lanes 0-15 hold K = 4s+0,4s+1; lanes 16-31 K = 4s+2,4s+3.
    float a[4][2];
    #pragma unroll
    for (int s = 0; s < 4; ++s) {
        #pragma unroll
        for (int j = 0; j < 2; ++j) {
            const int k = 4 * s + 2 * hi + j;
            a[s][j] = (k >= n) ? GLT.v[k - n] : 0.0f;
        }
    }
    // Carry scales gl^(16 - mrow) for this lane's 8 output rows (mrow = r + 8*hi).
    float cs[8];
    #pragma unroll
    for (int r = 0; r < 8; ++r)
        cs[r] = GLT.v[16 - (r + 8 * hi)];

    float carry = 0.0f;                        // adv at first row of later sub-tile
    #pragma unroll
    for (int st = 3; st >= 0; --st) {          // backward in time within the chunk
        const int t0 = chunk * CHUNK + st * 16;

        if (st > 0) {                          // prefetch next (earlier) sub-tile
            const size_t pb = (size_t)(t0 - 16 + 2 * hi) * BATCH + b0 + n;
            __builtin_prefetch(&reward[pb], 0, 3);   // high locality -> WGP scope
            __builtin_prefetch(&value[pb],  0, 3);
        }

        v8f c = {};
        #pragma unroll
        for (int s = 0; s < 4; ++s) {
            // B-matrix 4x16 layout: VGPR j holds K=j (lanes 0-15) / K=j+2 (16-31).
            const int    r0   = 4 * s + 2 * hi;
            const size_t base = (size_t)(t0 + r0) * BATCH + b0 + n;
            const float v0 = value[base];
            const float v1 = value[base + BATCH];
            const float v2 = value[base + 2 * (size_t)BATCH];
            const float d0 = reward[base]         + GAMMA * v1 - v0;
            const float d1 = reward[base + BATCH] + GAMMA * v2 - v1;
            v2f Am; Am.x = a[s][0]; Am.y = a[s][1];
            v2f Bm; Bm.x = d0;      Bm.y = d1;
            c = __builtin_amdgcn_wmma_f32_16x16x4_f32(
                    false, Am, false, Bm, (short)0, c, false, false);
        }

        // D layout: VGPR r -> (M=r, N=lane) lanes 0-15, (M=r+8, N=lane-16) 16-31.
        // st==3 (latest sub-tile): carry is identically zero, skip the FMA.
        float row0 = 0.0f;
        #pragma unroll
        for (int r = 0; r < 8; ++r) {
            const int   mrow = r + 8 * hi;
            const float adv  = (st == 3) ? c[r] : c[r] + cs[r] * carry;
            out[(size_t)(t0 + mrow) * BATCH + b0 + n] = adv;
            if (r == 0 && hi == 0) row0 = adv;   // adv[0][n] lives in lanes 0-15
        }
        carry = __shfl(row0, n, 32);             // broadcast adv[0][n] to both halves
    }

    if (hi == 0)
        heads[(size_t)chunk * BATCH + b0 + n] = carry;   // adv at t = chunk*64
}

// ---------------------------------------------------------------------------
// Pass 2: per-column scan over the 16 chunk heads; rewrites heads[] in place
// with carry_c = true adv at t=(c+1)*64 (0 for the last chunk).
// ---------------------------------------------------------------------------
__global__ __launch_bounds__(256) void gae_pass2(float* __restrict__ heads)
{
    const int b = (int)(blockIdx.x * blockDim.x + threadIdx.x);
    float A = 0.0f;
    #pragma unroll
    for (int c = NCHUNK - 1; c >= 0; --c) {
        const float h = heads[(size_t)c * BATCH + b];
        heads[(size_t)c * BATCH + b] = A;     // carry for chunk c
        A = h + GL64 * A;                     // true adv at t = c*64
    }
}

// ---------------------------------------------------------------------------
// Pass 3: fully parallel fix-up  adv[t][b] += gl^(64 - t%64) * carry_c[b],
// vectorized as float4 (B multiple of 4, rows contiguous). The table index is
// uniform per block (t is constant within a block of 256 threads).
// ---------------------------------------------------------------------------
__global__ __launch_bounds__(256) void gae_pass3(float* __restrict__ out,
                                                 const float* __restrict__ carry)
{
    const size_t idx = (size_t)blockIdx.x * blockDim.x + threadIdx.x; // T * B/4
    const int t  = (int)(idx >> 12);          // B/4 == 4096
    const int b4 = (int)(idx & 4095);
    const int ch = t >> 6;
    const float sc = GLT.v[CHUNK - (t & (CHUNK - 1))];

    const float4* c4 = (const float4*)carry;
    float4*       o4 = (float4*)out;
    float4 L = o4[(size_t)t  * 4096 + b4];
    const float4 K = c4[(size_t)ch * 4096 + b4];
    L.x += sc * K.x;  L.y += sc * K.y;  L.z += sc * K.z;  L.w += sc * K.w;
    o4[(size_t)t * 4096 + b4] = L;
}

extern "C" void kernel_launch(void* const* d_in, const int* in_sizes, int n_in,
                              void* d_out, int out_size, void* d_ws, size_t ws_size,
                              hipStream_t stream)
{
    const float* value  = (const float*)d_in[0];   // (T+1, B)
    const float* reward = (const float*)d_in[1];   // (T, B)
    float*       out    = (float*)d_out;           // (T, B)
    float*       heads  = (float*)d_ws;            // NCHUNK*BATCH floats = 1 MB

    // Pass 1: 16384 waves (one per 64t x 16b tile), 8 waves per block.
    gae_pass1<<<(NCHUNK * COLT) / 8, 256, 0, stream>>>(value, reward, out, heads);
    // Pass 2: one thread per column.
    gae_pass2<<<BATCH / 256, 256, 0, stream>>>(heads);
    // Pass 3: one thread per 4 elements.
    gae_pass3<<<(T * (BATCH / 4)) / 256, 256, 0, stream>>>(out, heads);
}